// SineGenerator_16844861735088
// MI455X (gfx1250) — compile-verified
//
#include <hip/hip_runtime.h>
#include <math.h>

#define HARM_DIM 9  // HARMONIC_NUM + 1

// Probe for the gfx1250 async global->LDS DMA builtins (device pass only; the
// host pass sees 0 which is fine because the macro only gates device code).
#if defined(__has_builtin)
#  if __has_builtin(__builtin_amdgcn_global_load_async_to_lds_b128)
#    define HAVE_ASYNC_LDS 1
#  endif
#endif
#ifndef HAVE_ASYNC_LDS
#  define HAVE_ASYNC_LDS 0
#endif

#if HAVE_ASYNC_LDS
// Builtin signature (from hipcc diagnostic): param 0 is a non-const
// 'int __vector(4)' pointer in address_space(1); param 1 mirrors it in
// address_space(3); params 2/3 are immediate offset / cache-policy ints.
typedef int v4i_t __attribute__((ext_vector_type(4)));
typedef __attribute__((address_space(1))) v4i_t* gvec_ptr;
typedef __attribute__((address_space(3))) v4i_t* lvec_ptr;
#endif

// ---------------------------------------------------------------------------
// Pass 2 (bandwidth-bound): one block per frame; 256 threads = 8 waves.
// Key simplification: with frame = b*L + l and Lo = L*upp,
//   elemOff = (b*Lo + l*upp)*9 = frame*upp*9   and   uvOff = frame*upp,
// so no integer division by L is needed at all. The async DMA issue loop is
// the first thing in the kernel so it overlaps the s_step/s_base setup.
// ---------------------------------------------------------------------------
__global__ void sine_synth(const float* __restrict__ noise_in,
                           const float* __restrict__ f0,
                           const float* __restrict__ stepArr,
                           const float* __restrict__ baseArr,
                           float* __restrict__ out_sine,
                           float* __restrict__ out_uv,
                           float* __restrict__ out_noise,
                           int upp) {
  extern __shared__ __align__(16) float tile[];  // upp * HARM_DIM floats
  __shared__ float s_step[HARM_DIM];
  __shared__ float s_base[HARM_DIM];

  const int frame = blockIdx.x;                       // == b*L + l
  const int nElem = upp * HARM_DIM;
  const long long elemOff = (long long)frame * nElem; // frame*upp*9

#if HAVE_ASYNC_LDS
  // Stage the frame's noise tile into LDS via CDNA5 async DMA (16 B / lane).
  const bool useLds = ((nElem & 3) == 0);
  if (useLds) {
    gvec_ptr gvec = (gvec_ptr)(noise_in + elemOff);   // 16B-aligned per frame
    lvec_ptr lvec = (lvec_ptr)(tile);
    const int nVec = nElem >> 2;
    for (int i = threadIdx.x; i < nVec; i += blockDim.x) {
      __builtin_amdgcn_global_load_async_to_lds_b128(gvec + i, lvec + i,
                                                     /*offset=*/0,
                                                     /*cpol=*/0);
    }
  }
#endif

  if (threadIdx.x < HARM_DIM) {
    int o = frame * HARM_DIM + threadIdx.x;
    s_step[threadIdx.x] = stepArr[o];
    s_base[threadIdx.x] = baseArr[o];
  }

  const float f0v = f0[frame];
  const float uvf = (f0v > 0.0f) ? 1.0f : 0.0f;
  const float amp = (f0v > 0.0f) ? 0.003f : (0.1f / 3.0f);

#if HAVE_ASYNC_LDS
  if (useLds) {
#  if __has_builtin(__builtin_amdgcn_s_wait_asynccnt)
    __builtin_amdgcn_s_wait_asynccnt(0);
#  else
    asm volatile("s_wait_asynccnt 0" ::: "memory");
#  endif
  }
#endif
  __syncthreads();  // publish s_step/s_base (+ LDS tile) to all 8 waves

  const float TWO_PI = 6.28318530717958647692f;
  for (int e = threadIdx.x; e < nElem; e += blockDim.x) {
    int k = e / HARM_DIM;                  // sample within frame (0..upp-1)
    int d = e - k * HARM_DIM;              // harmonic index
#if HAVE_ASYNC_LDS
    float n = useLds ? tile[e] : noise_in[elemOff + e];
#else
    float n = noise_in[elemOff + e];
#endif
    // phase(t) mod 1 == frac(base + (k+1)*step)   (cumsum_shift is integer)
    float ph = fmaf((float)(k + 1), s_step[d], s_base[d]);
    float fr = ph - floorf(ph);
    float s = sinf(TWO_PI * fr) * 0.1f;    // SINE_AMP
    float nz = amp * n;
    __builtin_nontemporal_store(fmaf(uvf, s, nz), &out_sine[elemOff + e]);
    __builtin_nontemporal_store(nz, &out_noise[elemOff + e]);
  }

  const long long uvOff = (long long)frame * upp;
  for (int k = threadIdx.x; k < upp; k += blockDim.x) {
    __builtin_nontemporal_store(uvf, &out_uv[uvOff + k]);
  }
}

// ---------------------------------------------------------------------------
// Pass 1 (tiny): per (batch, harmonic) sequential frame-rate scan.
//   step[b,l,d] = rad_values[b,l,d]            (incl. rand_ini at l==0)
//   base[b,l,d] = frac( upp * sum_{l'<l} rad ) (fp64 accumulator, mod-1 kept)
// 36 threads x 1000 iterations — microseconds.
// ---------------------------------------------------------------------------
__global__ void sine_frame_prefix(const float* __restrict__ f0,
                                  const float* __restrict__ rand_ini,
                                  float* __restrict__ stepArr,
                                  float* __restrict__ baseArr,
                                  int B, int L, int upp) {
  int id = blockIdx.x * blockDim.x + threadIdx.x;
  if (id >= B * HARM_DIM) return;
  int b = id / HARM_DIM;
  int d = id - b * HARM_DIM;

  const float mult = (float)(d + 1) * (1.0f / 48000.0f);  // (d+1)/SR
  const float ri = (d == 0) ? 0.0f : rand_ini[b * HARM_DIM + d];

  double acc = 0.0;                 // frac of upp * exclusive prefix sum
  const double uppd = (double)upp;
  for (int l = 0; l < L; ++l) {
    float rad = fmodf(f0[b * L + l] * mult, 1.0f);
    if (l == 0) rad += ri;          // reference adds rand_ini w/o re-mod
    int o = (b * L + l) * HARM_DIM + d;
    stepArr[o] = rad;
    baseArr[o] = (float)acc;        // exclusive prefix, already in [0,1)
    acc += uppd * (double)rad;
    acc -= floor(acc);
  }
}

// ---------------------------------------------------------------------------
// Host launcher. Inputs: f0 (B,L) f32, rand_ini (B,9) f32,
// noise_randn (B,Lo,9) f32, upp (scalar). All shapes recovered from in_sizes;
// upp = Lo / L so the device scalar never needs a sync-unsafe readback.
// Output layout: [sine (B*Lo*9) | uv (B*Lo) | noise (B*Lo*9)].
// ---------------------------------------------------------------------------
extern "C" void kernel_launch(void* const* d_in, const int* in_sizes, int n_in,
                              void* d_out, int out_size, void* d_ws,
                              size_t ws_size, hipStream_t stream) {
  const float* f0 = (const float*)d_in[0];
  const float* rand_ini = (const float*)d_in[1];
  const float* noise = (const float*)d_in[2];

  const int B = in_sizes[1] / HARM_DIM;                       // 4
  const int L = in_sizes[0] / B;                              // 1000
  const int Lo = (int)((long long)in_sizes[2] /
                       ((long long)B * HARM_DIM));            // 512000
  const int upp = Lo / L;                                     // 512

  // Workspace: step + base arrays, B*L*9 floats each (~144 KB each).
  float* stepArr = (float*)d_ws;
  float* baseArr = stepArr + (size_t)B * L * HARM_DIM;

  float* out_sine = (float*)d_out;
  float* out_uv = out_sine + (size_t)B * Lo * HARM_DIM;
  float* out_noise = out_uv + (size_t)B * Lo;

  const int nPref = B * HARM_DIM;
  sine_frame_prefix<<<(nPref + 63) / 64, 64, 0, stream>>>(
      f0, rand_ini, stepArr, baseArr, B, L, upp);

  const int nElem = upp * HARM_DIM;                           // 4608
  const size_t smem = (size_t)nElem * sizeof(float);          // 18 KB tile
  sine_synth<<<B * L, 256, smem, stream>>>(
      noise, f0, stepArr, baseArr, out_sine, out_uv, out_noise, upp);
}